// Loss_17729624998144
// MI455X (gfx1250) — compile-verified
//
#include <hip/hip_runtime.h>
#include <hip/hip_bf16.h>

typedef __attribute__((ext_vector_type(2))) float v2f;
typedef __attribute__((ext_vector_type(8))) float v8f;
// 4-float vector with only 4-byte alignment: lets clang emit global_load_b128
// for the contiguous bicubic row taps (CDNA5 supports unaligned VMEM).
typedef __attribute__((ext_vector_type(4), aligned(4))) float f4u;

#define H_DIM 1080
#define W_DIM 1920
#define PLANE (H_DIM * W_DIM)   // 2,073,600 = 8100 * 256
#define NCH 3

// ---------------------------------------------------------------------------
// Wave32 sum-reduction via V_WMMA_F32_16X16X4_F32.
// A (16x4 f32): VGPR0 = lanes 0-15 -> A[m][0], lanes 16-31 -> A[m][2];
// VGPR1 (K=1,3) zero.  B = all-ones 4x16.  D[m][n] = v(m) + v(m+16).
// Sum of the 8 D VGPRs per lane = half-sum of 16 rows; xor-16 add -> total.
// ---------------------------------------------------------------------------
__device__ __forceinline__ float wave_reduce_sum(float v) {
#if __has_builtin(__builtin_amdgcn_wmma_f32_16x16x4_f32)
    v2f a; a.x = v;    a.y = 0.0f;
    v2f b; b.x = 1.0f; b.y = 1.0f;
    v8f c = {};
    v8f d = __builtin_amdgcn_wmma_f32_16x16x4_f32(
        /*neg_a=*/false, a, /*neg_b=*/false, b,
        /*c_mod=*/(short)0, c, /*reuse_a=*/false, /*reuse_b=*/false);
    float s = d[0] + d[1] + d[2] + d[3] + d[4] + d[5] + d[6] + d[7];
    s += __shfl_xor(s, 16, 32);
    return s;
#else
    for (int off = 16; off > 0; off >>= 1) v += __shfl_xor(v, off, 32);
    return v;
#endif
}

// Catmull-Rom weights matching the reference's _cubic polynomial.
__device__ __forceinline__ void cubic_weights(float t, float w[4]) {
    float t2 = t * t, t3 = t2 * t;
    w[0] = 0.5f * (2.0f * t2 - t - t3);
    w[1] = 1.0f + 0.5f * (3.0f * t3 - 5.0f * t2);
    w[2] = 0.5f * (t + 4.0f * t2 - 3.0f * t3);
    w[3] = 0.5f * (t3 - t2);
}

// ---------------------------------------------------------------------------
// K1: flow-divergence threshold field  m0 = (|a+b| > 0.75); also zeroes acc.
//   a[y,x] = fx[y+1,x]-fx[y,x] (0 at last row)
//   b[y,x] = fy[y,x+1]-fy[y,x] (0 at last col)
// ---------------------------------------------------------------------------
__global__ void k_occ_field(const float* __restrict__ flow,
                            float* __restrict__ m0,
                            float* __restrict__ acc) {
    int idx = blockIdx.x * blockDim.x + threadIdx.x;
    if (idx == 0) acc[0] = 0.0f;
    if (idx >= PLANE) return;
    int y = idx / W_DIM;
    int x = idx - y * W_DIM;
    const float* fx = flow;          // channel 0
    const float* fy = flow + PLANE;  // channel 1
    float a = (y < H_DIM - 1) ? (fx[idx + W_DIM] - fx[idx]) : 0.0f;
    float b = (x < W_DIM - 1) ? (fy[idx + 1]     - fy[idx]) : 0.0f;
    m0[idx] = (fabsf(a + b) > 0.75f) ? 1.0f : 0.0f;
}

// ---------------------------------------------------------------------------
// K2: horizontal half of the separable 4x4 max-pool (dx in [-1,2], OOB = 0).
// ---------------------------------------------------------------------------
__global__ void k_dilate_h(const float* __restrict__ m0,
                           float* __restrict__ dil) {
    int idx = blockIdx.x * blockDim.x + threadIdx.x;
    if (idx >= PLANE) return;
    int y = idx / W_DIM;
    int x = idx - y * W_DIM;
    int row = y * W_DIM;
    float m = 0.0f;
    #pragma unroll
    for (int dx = -1; dx <= 2; ++dx) {
        int xx = x + dx;
        if (xx >= 0 && xx < W_DIM) m = fmaxf(m, m0[row + xx]);
    }
    dil[idx] = m;
}

// ---------------------------------------------------------------------------
// K3: fused main kernel.
//  - vertical half of max-pool + border -> occ
//  - bicubic footprint inside-test -> Mw = inside && !occ  (in {0,1})
//  - if Mw: 3-channel bicubic warps of input1/input2/fastdvdnet via b128 row
//    loads + the two L1 loss terms
//  - WMMA wave reduction, one atomicAdd per wave
// ---------------------------------------------------------------------------
__global__ void __launch_bounds__(256)
k_main_loss(const float* __restrict__ in1,
            const float* __restrict__ in2,
            const float* __restrict__ prev,
            const float* __restrict__ flow,
            const float* __restrict__ m10,
            const float* __restrict__ ex,
            const float* __restrict__ fast,
            const float* __restrict__ dil,
            float* __restrict__ acc) {
    int idx = blockIdx.x * blockDim.x + threadIdx.x;   // exact: PLANE = grid*256
    int y = idx / W_DIM;
    int x = idx - y * W_DIM;

    float local = 0.0f;

    // --- occlusion: vertical max over dy in [-1,2] of horizontally-dilated m0
    float dm = 0.0f;
    #pragma unroll
    for (int dy = -1; dy <= 2; ++dy) {
        int yy = y + dy;
        if (yy >= 0 && yy < H_DIM) dm = fmaxf(dm, dil[yy * W_DIM + x]);
    }
    bool occ = (dm > 0.5f) ||
               (y < 2) || (y >= H_DIM - 2) || (x < 2) || (x >= W_DIM - 2);

    // --- warp target + inside test
    float gx = (float)x + flow[idx];
    float gy = (float)y + flow[PLANE + idx];
    float fxf = floorf(gx);
    float fyf = floorf(gy);
    int x0 = (int)fxf - 1;
    int y0 = (int)fyf - 1;
    bool inside = (x0 >= 0) && (x0 + 3 <= W_DIM - 1) &&
                  (y0 >= 0) && (y0 + 3 <= H_DIM - 1);

    if (inside && !occ) {   // Mw == 1; otherwise both loss terms are 0
        float wx[4], wy[4];
        cubic_weights(gx - fxf, wx);
        cubic_weights(gy - fyf, wy);
        int ro[4];
        #pragma unroll
        for (int i = 0; i < 4; ++i) ro[i] = (y0 + i) * W_DIM + x0;

        #pragma unroll
        for (int c = 0; c < NCH; ++c) {
            const float* p1 = in1  + c * PLANE;
            const float* p2 = in2  + c * PLANE;
            const float* pf = fast + c * PLANE;
            float a1 = 0.0f, a2 = 0.0f, af = 0.0f;
            #pragma unroll
            for (int i = 0; i < 4; ++i) {
                float wyi = wy[i];
                int base = ro[i];
                f4u r1 = *(const f4u*)(p1 + base);   // one b128 per row
                f4u r2 = *(const f4u*)(p2 + base);
                f4u rf = *(const f4u*)(pf + base);
                #pragma unroll
                for (int j = 0; j < 4; ++j) {
                    float w = wyi * wx[j];
                    a1 = fmaf(w, r1[j], a1);
                    a2 = fmaf(w, r2[j], a2);
                    af = fmaf(w, rf[j], af);
                }
            }
            // single-use streams -> non-temporal (preserve L2 for gathers)
            float M1 = __builtin_nontemporal_load(m10  + c * PLANE + idx);
            float e  = __builtin_nontemporal_load(ex   + c * PLANE + idx);
            float pv = __builtin_nontemporal_load(prev + c * PLANE + idx);
            local += fabsf(e * M1 * a1 - e * M1 * pv);          // l1 term
            float g = e * (1.0f - M1);                          // * Mw (==1)
            local += fabsf(g * a2 - g * af);                    // l2 term
        }
    }

    // --- wave reduction via v_wmma (all lanes active: PLANE % 256 == 0)
    float wsum = wave_reduce_sum(local);
    if ((threadIdx.x & 31) == 0) atomicAdd(acc, wsum);
}

// ---------------------------------------------------------------------------
// K4: finalize  out = (sum1 + sum2) / (B*C*H*W)
// ---------------------------------------------------------------------------
__global__ void k_finalize(const float* __restrict__ acc,
                           float* __restrict__ out) {
    if (threadIdx.x == 0 && blockIdx.x == 0)
        out[0] = acc[0] * (1.0f / (float)(NCH * PLANE));
}

extern "C" void kernel_launch(void* const* d_in, const int* in_sizes, int n_in,
                              void* d_out, int out_size, void* d_ws, size_t ws_size,
                              hipStream_t stream) {
    const float* input1 = (const float*)d_in[0];
    const float* input2 = (const float*)d_in[1];
    const float* prev   = (const float*)d_in[2];
    const float* flow   = (const float*)d_in[3];
    const float* m10    = (const float*)d_in[4];
    const float* exm    = (const float*)d_in[5];
    const float* fast   = (const float*)d_in[6];
    float*       out    = (float*)d_out;

    float* wsf = (float*)d_ws;
    float* m0  = wsf;               // PLANE floats
    float* dil = wsf + PLANE;       // PLANE floats
    float* acc = wsf + 2 * PLANE;   // 1 float

    const int threads = 256;
    const int blocks  = (PLANE + threads - 1) / threads;   // 8100, exact fit

    k_occ_field<<<blocks, threads, 0, stream>>>(flow, m0, acc);
    k_dilate_h <<<blocks, threads, 0, stream>>>(m0, dil);
    k_main_loss<<<blocks, threads, 0, stream>>>(input1, input2, prev, flow,
                                                m10, exm, fast, dil, acc);
    k_finalize <<<1, 32, 0, stream>>>(acc, out);

    (void)in_sizes; (void)n_in; (void)out_size; (void)ws_size;
}